// Model_69767448756497
// MI455X (gfx1250) — compile-verified
//
#include <hip/hip_runtime.h>
#include <stdint.h>

// Scatter-add: out = self_tensor; out[linear_index[k]] += values[pos_idx[k]], D=64, f32.
// Pure bandwidth problem (~1.5 GB moved, 0.08 FLOP/byte) -> no WMMA; optimize the
// data-movement path: NT b128 streaming copy, async-to-LDS gather staging (ASYNCcnt),
// non-returning GLOBAL_ATOMIC_ADD_F32 scatter.

#define DD 64

typedef float vf4 __attribute__((ext_vector_type(4)));
typedef int   vi4 __attribute__((ext_vector_type(4)));
typedef __attribute__((address_space(1))) vi4* g_vi4_ptr;  // global int4*
typedef __attribute__((address_space(3))) vi4* l_vi4_ptr;  // LDS int4*

#if defined(__AMDGCN__) && __has_builtin(__builtin_amdgcn_global_load_async_to_lds_b128) && __has_builtin(__builtin_amdgcn_s_wait_asynccnt)
#define USE_ASYNC 1
#else
#define USE_ASYNC 0
#endif

// ---------------------------------------------------------------------------
// Kernel 1: streaming copy self_tensor -> out with non-temporal b128 ops.
// ---------------------------------------------------------------------------
__global__ __launch_bounds__(256) void copy_nt_kernel(const vf4* __restrict__ src,
                                                      vf4* __restrict__ dst,
                                                      long n4) {
    long i = (long)blockIdx.x * blockDim.x + threadIdx.x;
    const long stride = (long)gridDim.x * blockDim.x;
    for (; i < n4; i += stride) {
        vf4 v = __builtin_nontemporal_load(src + i);
        __builtin_nontemporal_store(v, dst + i);
    }
}

// ---------------------------------------------------------------------------
// Kernel 2: gather values[pos_idx[k]] (row of 64 f32) and scatter-add into
// out[linear_index[k]]. One wave handles 2 rows per iteration:
//   lane = (half = lane>>4, sub = lane&15); each lane owns 16B of one row.
// Async path: double-buffered GLOBAL_LOAD_ASYNC_TO_LDS_B128 stages next row
// pair while current pair's atomics are issued.
// ---------------------------------------------------------------------------
__global__ __launch_bounds__(256) void scatter_add_kernel(
    const float* __restrict__ values,
    const int*   __restrict__ lin_idx,
    const int*   __restrict__ pos_idx,
    float*       __restrict__ out,
    long K,
    const int*   __restrict__ accp) {

    __shared__ float lds[8 /*waves*/ * 2 /*bufs*/ * 128 /*floats = 2 rows*/];

    const int lane = threadIdx.x & 31;
    const int wib  = threadIdx.x >> 5;               // wave in block
    const int wpb  = blockDim.x >> 5;                // waves per block
    const long gw  = (long)blockIdx.x * wpb + wib;   // global wave id
    const long tw  = (long)gridDim.x * wpb;          // total waves
    const int half = lane >> 4;                      // which row of the pair
    const int sub  = lane & 15;                      // which float4 of the row
    const long P   = (K + 1) >> 1;                   // number of row pairs
    const int accumulate = accp[0];

    float* wbuf = &lds[wib * 256];                   // 2 bufs x 128 floats

#if USE_ASYNC
    auto issue = [&](long p, int b) {
        long k = 2 * p + half;
        if (k < K) {
            long pos = pos_idx[k];
            const float* g = values + pos * DD + sub * 4;
            float*       l = wbuf + b * 128 + half * 64 + sub * 4;
            __builtin_amdgcn_global_load_async_to_lds_b128(
                (g_vi4_ptr)(void*)g, (l_vi4_ptr)l, /*offset=*/0, /*cpol=*/0);
        }
    };

    long p = gw;
    if (p < P) issue(p, 0);
    int buf = 0;
    for (; p < P; p += tw, buf ^= 1) {
        long pn = p + tw;
        if (pn < P) {
            issue(pn, buf ^ 1);
            __builtin_amdgcn_s_wait_asynccnt(1);  // p's stage complete, pn in flight
        } else {
            __builtin_amdgcn_s_wait_asynccnt(0);
        }
        long k = 2 * p + half;
        if (k < K) {
            long li = lin_idx[k];
            float* dst = out + li * DD + sub * 4;
            const float* l = wbuf + buf * 128 + half * 64 + sub * 4;
            float a0 = l[0], a1 = l[1], a2 = l[2], a3 = l[3];
            if (accumulate) {
                atomicAdd(dst + 0, a0);
                atomicAdd(dst + 1, a1);
                atomicAdd(dst + 2, a2);
                atomicAdd(dst + 3, a3);
            } else {
                dst[0] = a0; dst[1] = a1; dst[2] = a2; dst[3] = a3;
            }
        }
    }
#else
    // Fallback: direct b128 gather with prefetch of the next row pair.
    for (long p = gw; p < P; p += tw) {
        long k = 2 * p + half;
        if (k < K) {
            long kn = 2 * (p + tw) + half;
            if (kn < K) {
                long posn = pos_idx[kn];
                __builtin_prefetch(values + posn * DD + sub * 4, 0, 0);
            }
            long pos = pos_idx[k];
            long li  = lin_idx[k];
            vf4 v = *(const vf4*)(values + pos * DD + sub * 4);
            float* dst = out + li * DD + sub * 4;
            if (accumulate) {
                atomicAdd(dst + 0, v.x);
                atomicAdd(dst + 1, v.y);
                atomicAdd(dst + 2, v.z);
                atomicAdd(dst + 3, v.w);
            } else {
                dst[0] = v.x; dst[1] = v.y; dst[2] = v.z; dst[3] = v.w;
            }
        }
    }
#endif
}

// ---------------------------------------------------------------------------
// Host-side launch.
// ---------------------------------------------------------------------------
extern "C" void kernel_launch(void* const* d_in, const int* in_sizes, int n_in,
                              void* d_out, int out_size, void* d_ws, size_t ws_size,
                              hipStream_t stream) {
    const float* self = (const float*)d_in[0];
    const int*   lin  = (const int*)  d_in[1];
    const int*   pos  = (const int*)  d_in[2];
    const float* vals = (const float*)d_in[3];
    // d_in[4] = slice_size (compile-time 64 here), d_in[5] = accumulate flag
    const int*   acc  = (const int*)  d_in[5];
    float*       out  = (float*)d_out;

    const long n  = (long)in_sizes[0];   // N * D
    const long K  = (long)in_sizes[1];   // number of scatter indices
    const long n4 = n >> 2;              // float4 count

    const int threads = 256;             // 8 wave32 per block
    const int cblocks = 2048;
    const int sblocks = 2048;

    copy_nt_kernel<<<cblocks, threads, 0, stream>>>((const vf4*)self, (vf4*)out, n4);
    scatter_add_kernel<<<sblocks, threads, 0, stream>>>(vals, lin, pos, out, K, acc);
}